// Mamba_82291573391740
// MI455X (gfx1250) — compile-verified
//
#include <hip/hip_runtime.h>
#include <hip/hip_bf16.h>
#include <cstddef>
#include <cstdint>

// Mamba block forward, fp32, MI455X (gfx1250).
// GEMMs: V_WMMA_F32_16X16X4_F32, 128x128 block tile, 32x64 per-wave tile,
// statically double-buffered LDS filled by GLOBAL_LOAD_ASYNC_TO_LDS_B128.

#define D_MODEL_  2048
#define D_STATE_  16
#define D_CONV_   4
#define D_INNER_  4096
#define DT_RANK_  128
#define BATCH_    2
#define SEQLEN_   2048
#define MROWS_    (BATCH_*SEQLEN_)        // 4096
#define XDBL_W_   (DT_RANK_+2*D_STATE_)   // 160

typedef float v2f __attribute__((ext_vector_type(2)));
typedef float v8f __attribute__((ext_vector_type(8)));

__device__ __forceinline__ float sigmoidf_(float x){ return 1.0f/(1.0f+__expf(-x)); }
__device__ __forceinline__ float siluf_(float x){ return x*sigmoidf_(x); }
__device__ __forceinline__ float softplusf_(float x){ return (x>20.0f)?x:log1pf(__expf(x)); }

#define BK_  16   // K chunk
#define LDT_ 20   // LDS row stride (dwords): 16B-aligned rows, conflict-free frag reads
#define BM_  128
#define BN_  128

// one lane: copy 16 bytes global -> LDS via async DMA (GVS: SGPR base + u32 offset)
__device__ __forceinline__ void async_b128(uint32_t lds_byte, const float* base, uint32_t g_byte) {
  asm volatile("global_load_async_to_lds_b128 %0, %1, %2"
               :: "v"(lds_byte), "v"(g_byte), "s"(base) : "memory");
}
__device__ __forceinline__ void wait_async0() {
  asm volatile("s_wait_asynccnt 0x0" ::: "memory");
}
__device__ __forceinline__ uint32_t lds_addr(const void* p) {
  return (uint32_t)(uintptr_t)p;   // low 32 bits of flat LDS address = LDS offset
}

// C[M,N] = A[M,K] * W[N,K]^T.  A row stride lda, C row stride ldc, W stride K.
// mode 0: store C; mode 1: store softplus(C + bias[n]).
// Requires M%128==0, K%16==0; N edge handled by clamp+store-guard.
__global__ __launch_bounds__(256) void gemm_wmma_f32(
    const float* __restrict__ A, int lda,
    const float* __restrict__ W,
    float* __restrict__ C, int ldc,
    int M, int N, int K,
    const float* __restrict__ bias, int mode)
{
  __shared__ __align__(16) float sA[2*BM_*LDT_];
  __shared__ __align__(16) float sB[2*BN_*LDT_];
  const int m0 = blockIdx.y * BM_;
  const int n0 = blockIdx.x * BN_;
  const int tid  = threadIdx.x;
  const int lane = tid & 31;
  const int wave = tid >> 5;
  const int mW   = (wave >> 1) * 32;   // waves: 4 in M x 2 in N
  const int nW   = (wave & 1) * 64;    // each wave owns 32x64 of C

  // per-thread DMA assignment: rows r and r+64 of both tiles, 16B each
  const int r  = tid >> 2;             // 0..63
  const int kq = (tid & 3) * 4;        // 0,4,8,12
  const int nr1 = (n0 + r      < N) ? (n0 + r)      : (N - 1);  // clamp keeps loads
  const int nr2 = (n0 + r + 64 < N) ? (n0 + r + 64) : (N - 1);  // in bounds; cols guarded
  const uint32_t aL1[2] = { lds_addr(&sA[ r     *LDT_+kq]), lds_addr(&sA[BM_*LDT_ +  r     *LDT_+kq]) };
  const uint32_t aL2[2] = { lds_addr(&sA[(r+64)*LDT_+kq]), lds_addr(&sA[BM_*LDT_ + (r+64)*LDT_+kq]) };
  const uint32_t bL1[2] = { lds_addr(&sB[ r     *LDT_+kq]), lds_addr(&sB[BN_*LDT_ +  r     *LDT_+kq]) };
  const uint32_t bL2[2] = { lds_addr(&sB[(r+64)*LDT_+kq]), lds_addr(&sB[BN_*LDT_ + (r+64)*LDT_+kq]) };
  const uint32_t aO1 = (uint32_t)(((size_t)(m0 + r     )*lda + kq) * 4);
  const uint32_t aO2 = (uint32_t)(((size_t)(m0 + r + 64)*lda + kq) * 4);
  const uint32_t bO1 = (uint32_t)(((size_t)nr1*K + kq) * 4);
  const uint32_t bO2 = (uint32_t)(((size_t)nr2*K + kq) * 4);

  v8f acc[2][4] = {};

  auto issue = [&](int chunk, int sel) {
    const uint32_t kb = (uint32_t)(chunk * BK_) * 4u;
    async_b128(aL1[sel], A, aO1 + kb);
    async_b128(aL2[sel], A, aO2 + kb);
    async_b128(bL1[sel], W, bO1 + kb);
    async_b128(bL2[sel], W, bO2 + kb);
  };

  // fragment gathers per ISA 7.12.2 fp32 layout:
  // lanes 0-15: row lane&15, K={kk,kk+1}; lanes 16-31: same rows, K={kk+2,kk+3}
  auto compute = [&](int sel) {
    const int rA = sel*BM_*LDT_ + (mW + (lane & 15))*LDT_;
    const int rB = sel*BN_*LDT_ + (nW + (lane & 15))*LDT_;
    const int kh = (lane >> 4) << 1;
#pragma unroll
    for (int kk = 0; kk < BK_; kk += 4) {
      v2f a0, a1, b[4];
      a0.x = sA[rA           + kk + kh]; a0.y = sA[rA           + kk + kh + 1];
      a1.x = sA[rA + 16*LDT_ + kk + kh]; a1.y = sA[rA + 16*LDT_ + kk + kh + 1];
#pragma unroll
      for (int q = 0; q < 4; ++q) {
        b[q].x = sB[rB + q*16*LDT_ + kk + kh];
        b[q].y = sB[rB + q*16*LDT_ + kk + kh + 1];
      }
#pragma unroll
      for (int q = 0; q < 4; ++q) {
        acc[0][q] = __builtin_amdgcn_wmma_f32_16x16x4_f32(false, a0, false, b[q],
                                                          (short)0, acc[0][q], false, false);
        acc[1][q] = __builtin_amdgcn_wmma_f32_16x16x4_f32(false, a1, false, b[q],
                                                          (short)0, acc[1][q], false, false);
      }
    }
  };

  const int NC = K / BK_;
  issue(0, 0);
  wait_async0();
  __syncthreads();
  for (int c = 0; c < NC; c += 2) {            // 2x unroll: static buffer selection
    if (c + 1 < NC) issue(c + 1, 1);
    compute(0);
    wait_async0();
    __syncthreads();
    if (c + 1 >= NC) break;
    if (c + 2 < NC) issue(c + 2, 0);
    compute(1);
    wait_async0();
    __syncthreads();
  }

  // D layout: VGPR j -> row j + 8*(lane>=16), col = lane&15
  const int cb = n0 + nW + (lane & 15);
#pragma unroll
  for (int mi = 0; mi < 2; ++mi) {
    const int rowB = m0 + mW + mi*16 + ((lane >> 4) << 3);
#pragma unroll
    for (int j = 0; j < 8; ++j) {
      const int rr = rowB + j;
#pragma unroll
      for (int q = 0; q < 4; ++q) {
        const int cc = cb + q*16;
        if (cc < N) {
          float v = acc[mi][q][j];
          if (mode == 1) v = softplusf_(v + bias[cc]);
          C[(size_t)rr*ldc + cc] = v;
        }
      }
    }
  }
}

// Depthwise causal conv1d (width 4) + SiLU; x = first half of xz rows.
__global__ __launch_bounds__(256) void conv_silu_kernel(
    const float* __restrict__ xz, const float* __restrict__ cw,
    const float* __restrict__ cb, float* __restrict__ xc)
{
  const int idx = blockIdx.x*256 + threadIdx.x;
  if (idx >= MROWS_*D_INNER_) return;
  const int d = idx & (D_INNER_-1);
  const int l = (idx >> 12) & (SEQLEN_-1);
  const int b = idx >> 23;
  const size_t rs = (size_t)(2*D_INNER_);
  const float* xrow = xz + (size_t)(b*SEQLEN_)*rs + d;
  float acc = cb[d];
#pragma unroll
  for (int j = 0; j < D_CONV_; ++j) {
    const int ll = l - (D_CONV_-1) + j;
    if (ll >= 0) acc += xrow[(size_t)ll*rs] * cw[d*D_CONV_ + j];
  }
  xc[idx] = siluf_(acc);
}

// Selective scan: one thread per (b,d), 16-state h + A row in regs.
// B/C staged 32 timesteps at a time in LDS (2 barriers / 32 steps).
__global__ __launch_bounds__(256) void scan_kernel(
    const float* __restrict__ xdbl, const float* __restrict__ delta,
    const float* __restrict__ xc, const float* __restrict__ A_log,
    float* __restrict__ y)
{
  __shared__ __align__(16) float sBC[32][32];   // [t][0..15]=B, [16..31]=C
  const int b = blockIdx.x / (D_INNER_/256);
  const int d = (blockIdx.x % (D_INNER_/256))*256 + threadIdx.x;
  float h[D_STATE_], Arow[D_STATE_];
#pragma unroll
  for (int n = 0; n < D_STATE_; ++n) {
    h[n] = 0.f;
    Arow[n] = -__expf(A_log[d*D_STATE_ + n]);
  }
  const int jt = (threadIdx.x*4) >> 5;     // timestep this thread stages
  const int jc = (threadIdx.x*4) & 31;     // 4 contiguous B/C entries
  for (int l0 = 0; l0 < SEQLEN_; l0 += 32) {
    const float4 v = *(const float4*)&xdbl[
        (size_t)(b*SEQLEN_ + l0 + jt)*XDBL_W_ + DT_RANK_ + jc];
    *(float4*)&sBC[jt][jc] = v;
    __syncthreads();
#pragma unroll 4
    for (int t = 0; t < 32; ++t) {
      const size_t off = (size_t)(b*SEQLEN_ + l0 + t)*D_INNER_ + d;
      const float dt = delta[off];
      const float du = dt * xc[off];
      float yv = 0.f;
#pragma unroll
      for (int n = 0; n < D_STATE_; ++n) {
        h[n] = h[n]*__expf(dt*Arow[n]) + du*sBC[t][n];
        yv  += h[n]*sBC[t][D_STATE_ + n];
      }
      y[off] = yv;
    }
    __syncthreads();
  }
}

// y = (y + D[d]*xc) * silu(z); z = second half of xz rows
__global__ __launch_bounds__(256) void gate_kernel(
    float* __restrict__ y, const float* __restrict__ xc,
    const float* __restrict__ Dv, const float* __restrict__ xz)
{
  const int idx = blockIdx.x*256 + threadIdx.x;
  if (idx >= MROWS_*D_INNER_) return;
  const int d = idx & (D_INNER_-1);
  const size_t row = (size_t)(idx >> 12);   // b*L + l
  const float z = xz[row*(size_t)(2*D_INNER_) + D_INNER_ + d];
  y[idx] = (y[idx] + Dv[d]*xc[idx]) * siluf_(z);
}

extern "C" void kernel_launch(void* const* d_in, const int* in_sizes, int n_in,
                              void* d_out, int out_size, void* d_ws, size_t ws_size,
                              hipStream_t stream) {
  (void)in_sizes; (void)n_in; (void)out_size; (void)ws_size;
  const float* hidden     = (const float*)d_in[0];
  const float* in_proj_w  = (const float*)d_in[1];
  const float* conv_w     = (const float*)d_in[2];
  const float* conv_b     = (const float*)d_in[3];
  const float* x_proj_w   = (const float*)d_in[4];
  const float* dt_proj_w  = (const float*)d_in[5];
  const float* dt_proj_b  = (const float*)d_in[6];
  const float* A_log      = (const float*)d_in[7];
  const float* Dvec       = (const float*)d_in[8];
  const float* out_proj_w = (const float*)d_in[9];
  float* out = (float*)d_out;

  float* ws    = (float*)d_ws;
  float* xz    = ws;                                    // 4096 x 8192
  float* xc    = xz   + (size_t)MROWS_*2*D_INNER_;      // 4096 x 4096
  float* xdbl  = xc   + (size_t)MROWS_*D_INNER_;        // 4096 x 160
  float* delta = xdbl + (size_t)MROWS_*XDBL_W_;         // 4096 x 4096
  float* ybuf  = delta+ (size_t)MROWS_*D_INNER_;        // 4096 x 4096

  const dim3 blk(256);
  const int nElem = MROWS_*D_INNER_;

  // 1) in_proj: xz = hidden * in_proj_w^T        (4096 x 8192, K=2048)
  gemm_wmma_f32<<<dim3(2*D_INNER_/BN_, MROWS_/BM_), blk, 0, stream>>>(
      hidden, D_MODEL_, in_proj_w, xz, 2*D_INNER_,
      MROWS_, 2*D_INNER_, D_MODEL_, nullptr, 0);

  // 2) depthwise conv + SiLU -> xc
  conv_silu_kernel<<<(nElem+255)/256, blk, 0, stream>>>(xz, conv_w, conv_b, xc);

  // 3) x_proj: xdbl = xc * x_proj_w^T            (4096 x 160, K=4096)
  gemm_wmma_f32<<<dim3((XDBL_W_+BN_-1)/BN_, MROWS_/BM_), blk, 0, stream>>>(
      xc, D_INNER_, x_proj_w, xdbl, XDBL_W_,
      MROWS_, XDBL_W_, D_INNER_, nullptr, 0);

  // 4) dt_proj + bias + softplus: delta          (4096 x 4096, K=128, lda=160)
  gemm_wmma_f32<<<dim3(D_INNER_/BN_, MROWS_/BM_), blk, 0, stream>>>(
      xdbl, XDBL_W_, dt_proj_w, delta, D_INNER_,
      MROWS_, D_INNER_, DT_RANK_, dt_proj_b, 1);

  // 5) selective scan -> ybuf
  scan_kernel<<<BATCH_*(D_INNER_/256), blk, 0, stream>>>(
      xdbl, delta, xc, A_log, ybuf);

  // 6) gate: ybuf = (ybuf + D*xc) * silu(z)
  gate_kernel<<<(nElem+255)/256, blk, 0, stream>>>(ybuf, xc, Dvec, xz);

  // 7) out_proj: out = ybuf * out_proj_w^T       (4096 x 2048, K=4096)
  gemm_wmma_f32<<<dim3(D_MODEL_/BN_, MROWS_/BM_), blk, 0, stream>>>(
      ybuf, D_INNER_, out_proj_w, out, D_MODEL_,
      MROWS_, D_MODEL_, D_INNER_, nullptr, 0);
}